// HandGSConsistencyLoss_58128087384445
// MI455X (gfx1250) — compile-verified
//
#include <hip/hip_runtime.h>
#include <hip/hip_bf16.h>

typedef __attribute__((ext_vector_type(2))) float v2f;
typedef __attribute__((ext_vector_type(8))) float v8f;

#define BS        64        // B*S
#define HW        65536
#define NV        778
#define MESHPAD   1568      // 2*778=1556 padded to 98*16
#define KSEL      1024
#define INF_DIST  1.0e6f

// ---------------- init: sentinel-fill workspace ----------------
__global__ __launch_bounds__(256) void init_kernel(
    v2f* mesh_bxy, v2f* mesh_bz0, float* sbv,
    v2f* samp_axy, v2f* samp_az0, float* sa,
    unsigned* T, int* selcnt, float* gsum)
{
    int i = blockIdx.x * blockDim.x + threadIdx.x;
    const int nm = BS * MESHPAD;
    if (i < nm) {
        mesh_bxy[i] = (v2f){INF_DIST, INF_DIST};
        mesh_bz0[i] = (v2f){INF_DIST, 0.0f};
        sbv[i] = 3.0f * INF_DIST * INF_DIST;
    }
    if (i < BS * KSEL) {
        samp_axy[i] = (v2f){0.f, 0.f};
        samp_az0[i] = (v2f){0.f, 0.f};
        sa[i] = 0.f;
    }
    if (i < BS) { T[i] = 0u; selcnt[i] = 0; }
    if (i == 0) gsum[0] = 0.f;
}

// ---------------- MANO point generation ----------------
__global__ __launch_bounds__(128) void mano_kernel(
    const float* __restrict__ hand_params, const int* __restrict__ hand_valid,
    const float* __restrict__ vt_l, const float* __restrict__ vt_r,
    const float* __restrict__ sd_l, const float* __restrict__ sd_r,
    const float* __restrict__ pd_l, const float* __restrict__ pd_r,
    v2f* __restrict__ mesh_bxy, v2f* __restrict__ mesh_bz0,
    float* __restrict__ sbv)
{
    const int f = blockIdx.x;     // 0..63
    const int h = blockIdx.y;     // 0..1
    __shared__ float p[32];
    if (threadIdx.x < 32) p[threadIdx.x] = hand_params[f * 64 + h * 32 + threadIdx.x];
    __syncthreads();

    const int valid = hand_valid[f * 2 + h];
    const float* vt = h ? vt_r : vt_l;
    const float* sd = h ? sd_r : sd_l;
    const float* pd = h ? pd_r : pd_l;

    // quat (wxyz) -> rotvec, exactly mirroring the reference
    float qw = p[3], qx = p[4], qy = p[5], qz = p[6];
    float qn = sqrtf(qw*qw + qx*qx + qy*qy + qz*qz + 1e-8f);
    float inv = 1.0f / fmaxf(qn, 1e-8f);
    qw *= inv; qx *= inv; qy *= inv; qz *= inv;
    float sgn = (qw < 0.f) ? -1.f : 1.f;
    qw *= sgn; qx *= sgn; qy *= sgn; qz *= sgn;
    float sh  = sqrtf(qx*qx + qy*qy + qz*qz + 1e-8f);
    float wsf = fminf(fmaxf(qw, -1.f + 1e-8f), 1.f - 1e-8f);
    float ang = 2.f * atan2f(sh, wsf);
    float fac = (sh < 1e-6f) ? 2.f : (ang / fmaxf(sh, 1e-8f));
    float rvx = qx * fac, rvy = qy * fac, rvz = qz * fac;

    // Rodrigues constants
    float angr = sqrtf(rvx*rvx + rvy*rvy + rvz*rvz + 1e-8f);
    float kx = rvx / angr, ky = rvy / angr, kz = rvz / angr;
    float cA = cosf(angr), sA = sinf(angr), oc = 1.f - cA;
    float tx = p[0], ty = p[1], tz = p[2];

    for (int v = threadIdx.x; v < NV; v += blockDim.x) {
        float px = vt[v*3+0], py = vt[v*3+1], pz = vt[v*3+2];
        #pragma unroll
        for (int k = 0; k < 10; ++k) {
            float bk = p[22 + k];
            px += sd[(v*3+0)*10 + k] * bk;
            py += sd[(v*3+1)*10 + k] * bk;
            pz += sd[(v*3+2)*10 + k] * bk;
        }
        #pragma unroll
        for (int k = 0; k < 15; ++k) {
            float hk = p[7 + k];
            px += pd[(v*3+0)*15 + k] * hk;
            py += pd[(v*3+1)*15 + k] * hk;
            pz += pd[(v*3+2)*15 + k] * hk;
        }
        float cxx = ky*pz - kz*py, cyy = kz*px - kx*pz, czz = kx*py - ky*px;
        float kdv = kx*px + ky*py + kz*pz;
        float ox = px*cA + cxx*sA + kx*kdv*oc + tx;
        float oy = py*cA + cyy*sA + ky*kdv*oc + ty;
        float oz = pz*cA + czz*sA + kz*kdv*oc + tz;
        if (!valid) { ox = INF_DIST; oy = INF_DIST; oz = INF_DIST; }
        int idx = f * MESHPAD + h * NV + v;
        mesh_bxy[idx] = (v2f){ox, oy};        // B operand K0/K1 pair
        mesh_bz0[idx] = (v2f){oz, 0.0f};      // B operand K2/K3 pair
        sbv[idx] = ox*ox + oy*oy + oz*oz;
    }
}

// ---------------- exact top-k threshold via 3-level radix select ----------------
__global__ __launch_bounds__(256) void topk_thresh_kernel(
    const float* __restrict__ mask, unsigned* __restrict__ T)
{
    const int f = blockIdx.x;
    __shared__ unsigned hist[2048];
    __shared__ unsigned s_prefix, s_plen, s_krem;
    if (threadIdx.x == 0) { s_prefix = 0u; s_plen = 0u; s_krem = KSEL; }
    __syncthreads();

    const unsigned widths[3] = {11u, 11u, 10u};
    for (int lev = 0; lev < 3; ++lev) {
        const unsigned width = widths[lev];
        for (int i = threadIdx.x; i < 2048; i += 256) hist[i] = 0u;
        __syncthreads();
        const unsigned prefix = s_prefix, plen = s_plen;
        const unsigned shift = 32u - plen - width;
        const unsigned bmask = (1u << width) - 1u;
        for (int i = threadIdx.x; i < HW; i += 256) {
            unsigned u = __float_as_uint(mask[(long)f * HW + i]);
            bool match = (plen == 0u) || ((u >> (32u - plen)) == prefix);
            if (match) atomicAdd(&hist[(u >> shift) & bmask], 1u);
        }
        __syncthreads();
        if (threadIdx.x == 0) {
            unsigned krem = s_krem, chosen = 0u;
            for (int b = (int)bmask; b >= 0; --b) {
                unsigned c = hist[b];
                if (krem <= c) { chosen = (unsigned)b; break; }
                krem -= c;
            }
            s_krem = krem;
            s_prefix = (prefix << width) | chosen;
            s_plen = plen + width;
        }
        __syncthreads();
    }
    if (threadIdx.x == 0) T[f] = s_prefix;   // exact bit pattern of k-th largest
}

// ---------------- compact selected samples (A operand pre-scaled by -2) ----------------
__global__ __launch_bounds__(256) void gather_kernel(
    const float* __restrict__ mask, const float* __restrict__ means,
    const int* __restrict__ hand_valid, const unsigned* __restrict__ T,
    int* __restrict__ selcnt,
    v2f* __restrict__ samp_axy, v2f* __restrict__ samp_az0,
    float* __restrict__ sa)
{
    const int f = blockIdx.y;
    const int anyhv = hand_valid[f*2+0] || hand_valid[f*2+1];
    if (!anyhv) return;
    const unsigned t = T[f];
    const int base = blockIdx.x * 4096;
    for (int i = base + threadIdx.x; i < base + 4096; i += 256) {
        float v = mask[(long)f * HW + i];
        if (v > 0.5f && __float_as_uint(v) >= t) {
            int slot = atomicAdd(&selcnt[f], 1);
            if (slot < KSEL) {
                long off = ((long)f * HW + i) * 3;
                float x = means[off+0], y = means[off+1], z = means[off+2];
                int o = f * KSEL + slot;
                samp_axy[o] = (v2f){-2.0f * x, -2.0f * y};
                samp_az0[o] = (v2f){-2.0f * z, 0.0f};
                sa[o] = x*x + y*y + z*z;
            }
        }
    }
}

// ---------------- WMMA nearest-neighbor + accumulate ----------------
// d2 = sa + sbv + wmma( (-2*S), M^T ),  K=3 padded to 4 (f32 WMMA 16x16x4).
// A layout (ISA 7.12.2): lane<16 -> row=lane, K0/K1 = (-2x,-2y); lane>=16 -> same row, K2/K3 = (-2z,0).
// B layout mirrors it for columns; C layout: VGPR j -> M=j (lanes 0-15), M=j+8 (lanes 16-31), N=lane%16.
// All operand fetch is branch-free: each lane picks its packed-pair base pointer once.
__global__ __launch_bounds__(256) void nn_wmma_kernel(
    const v2f* __restrict__ mesh_bxy, const v2f* __restrict__ mesh_bz0,
    const float* __restrict__ sbv,
    const v2f* __restrict__ samp_axy, const v2f* __restrict__ samp_az0,
    const float* __restrict__ sa,
    const int* __restrict__ selcnt, float* __restrict__ gsum)
{
    const int f    = blockIdx.x;                 // frame
    const int wave = (int)(threadIdx.x >> 5);    // 8 waves/block
    const int lane = (int)(threadIdx.x & 31);
    const int tile = blockIdx.y * 8 + wave;      // 0..63 sample tiles
    const int half = lane >> 4;
    const int l16  = lane & 15;

    int n = selcnt[f]; if (n > KSEL) n = KSEL;

    // Lane-selected base pointers (uniform per half-wave, chosen once).
    const v2f* pa = (half ? samp_az0 : samp_axy) + (size_t)f * KSEL;
    const v2f* pb = (half ? mesh_bz0 : mesh_bxy) + (size_t)f * MESHPAD;
    const float* fsb = sbv + (size_t)f * MESHPAD;

    const int row = tile * 16 + l16;
    const v2f a = pa[row];

    float tmin[8];
    #pragma unroll
    for (int j = 0; j < 8; ++j) tmin[j] = 3.0e38f;

    #pragma unroll 2
    for (int mt = 0; mt < MESHPAD / 16; ++mt) {
        const int col = mt * 16 + l16;
        const v2f   b   = pb[col];     // single global_load_b64, branch-free
        const float sbl = fsb[col];
        v8f c = {};
        c = __builtin_amdgcn_wmma_f32_16x16x4_f32(
                /*neg_a=*/false, a, /*neg_b=*/false, b,
                /*c_mod=*/(short)0, c, /*reuse_a=*/false, /*reuse_b=*/false);
        #pragma unroll
        for (int j = 0; j < 8; ++j) {
            float val = sbl + c[j];                 // ||m||^2 - 2 s.m
            tmin[j] = fminf(tmin[j], val);
        }
    }

    // butterfly min across the 16 lanes of each half (N dimension)
    #pragma unroll
    for (int m = 1; m < 16; m <<= 1) {
        #pragma unroll
        for (int j = 0; j < 8; ++j)
            tmin[j] = fminf(tmin[j], __shfl_xor(tmin[j], m, 32));
    }

    if (l16 == 0) {
        float local = 0.f;
        const float* fsa = sa + (size_t)f * KSEL;
        #pragma unroll
        for (int j = 0; j < 8; ++j) {
            int M = tile * 16 + half * 8 + j;
            if (M < n) local += fmaxf(fsa[M] + tmin[j], 0.f);
        }
        atomicAdd(gsum, local);
    }
}

// ---------------- finalize ----------------
__global__ void final_kernel(const float* gsum, const int* selcnt, float* out)
{
    if (threadIdx.x == 0 && blockIdx.x == 0) {
        int cnt = 0;
        for (int f = 0; f < BS; ++f) { int c = selcnt[f]; if (c > KSEL) c = KSEL; cnt += c; }
        out[0] = gsum[0] / fmaxf((float)cnt, 1.0f);
    }
}

extern "C" void kernel_launch(void* const* d_in, const int* in_sizes, int n_in,
                              void* d_out, int out_size, void* d_ws, size_t ws_size,
                              hipStream_t stream)
{
    const float* hand_params = (const float*)d_in[0];
    const float* gs_means    = (const float*)d_in[1];
    const float* gs_mask     = (const float*)d_in[2];
    const int*   hand_valid  = (const int*)  d_in[3];
    const float* vt_l = (const float*)d_in[4];
    const float* vt_r = (const float*)d_in[5];
    const float* sd_l = (const float*)d_in[6];
    const float* sd_r = (const float*)d_in[7];
    const float* pd_l = (const float*)d_in[8];
    const float* pd_r = (const float*)d_in[9];

    float* ws = (float*)d_ws;
    const size_t NM = (size_t)BS * MESHPAD;   // 100352
    const size_t NS = (size_t)BS * KSEL;      // 65536
    v2f*      mesh_bxy = (v2f*)ws;                       // 2*NM floats
    v2f*      mesh_bz0 = (v2f*)(ws + 2 * NM);            // 2*NM floats
    float*    sbv      = ws + 4 * NM;                    // NM floats
    v2f*      samp_axy = (v2f*)(ws + 5 * NM);            // 2*NS floats
    v2f*      samp_az0 = (v2f*)(ws + 5 * NM + 2 * NS);   // 2*NS floats
    float*    sa       = ws + 5 * NM + 4 * NS;           // NS floats
    unsigned* T        = (unsigned*)(ws + 5 * NM + 5 * NS);
    int*      selcnt   = (int*)(T + BS);
    float*    gsum     = (float*)(selcnt + BS);

    init_kernel<<<(int)((NM + 255) / 256), 256, 0, stream>>>(
        mesh_bxy, mesh_bz0, sbv, samp_axy, samp_az0, sa, T, selcnt, gsum);

    mano_kernel<<<dim3(BS, 2), 128, 0, stream>>>(
        hand_params, hand_valid, vt_l, vt_r, sd_l, sd_r, pd_l, pd_r,
        mesh_bxy, mesh_bz0, sbv);

    topk_thresh_kernel<<<BS, 256, 0, stream>>>(gs_mask, T);

    gather_kernel<<<dim3(HW / 4096, BS), 256, 0, stream>>>(
        gs_mask, gs_means, hand_valid, T, selcnt, samp_axy, samp_az0, sa);

    nn_wmma_kernel<<<dim3(BS, 8), 256, 0, stream>>>(
        mesh_bxy, mesh_bz0, sbv, samp_axy, samp_az0, sa, selcnt, gsum);

    final_kernel<<<1, 32, 0, stream>>>(gsum, selcnt, (float*)d_out);
}